// pos_encoding_22016002360039
// MI455X (gfx1250) — compile-verified
//
#include <hip/hip_runtime.h>
#include <math.h>

// out[b, s, d] = x[b, s, d] + pe[s, d]
//   pe[s, d] = (d even) ? sin(s / 10000^(2d/D)) : cos(s / 10000^(2d/D))
// x: fp32 [B=8, S=4096, D=1024]  -> pure HBM-bandwidth problem (~268 MB traffic).
// Strategy: 1 block per row s; pe computed once in registers, reused across all
// 8 batches; x streamed via CDNA5 async global->LDS double-buffered pipeline;
// output written with non-temporal stores.

typedef float v4f __attribute__((ext_vector_type(4)));
typedef int   v4i __attribute__((vector_size(16)));   // matches builtin's param type

#define PES_S   4096
#define PES_D   1024
#define THREADS 256          // 256 threads * float4 = 1024 columns = one row

#if defined(__has_builtin)
#  if __has_builtin(__builtin_amdgcn_global_load_async_to_lds_b128)
#    define HAVE_ASYNC_LDS 1
#  endif
#  if __has_builtin(__builtin_amdgcn_s_wait_asynccnt)
#    define HAVE_WAIT_ASYNC 1
#  endif
#endif

// Address-space-qualified typed pointers for the async builtin. If the builtin
// declares AS-qualified params this is an exact match; if it declares generic
// params, AS->generic is an implicit (superset) conversion. Both compile.
typedef __attribute__((address_space(1))) v4i* g4ptr;
typedef __attribute__((address_space(3))) v4i* l4ptr;

template <int N>
__device__ __forceinline__ void wait_asynccnt_le() {
#if defined(HAVE_WAIT_ASYNC)
  __builtin_amdgcn_s_wait_asynccnt(N);
#elif defined(HAVE_ASYNC_LDS)
  if constexpr (N == 0) {
    asm volatile("s_wait_asynccnt 0" ::: "memory");
  } else {
    asm volatile("s_wait_asynccnt 1" ::: "memory");
  }
#endif
}

__global__ __launch_bounds__(THREADS) void pe_add_kernel(
    const float* __restrict__ x, float* __restrict__ out, int B) {
  const int s   = blockIdx.x;        // sequence position, 0..S-1
  const int tid = threadIdx.x;       // 0..255
  const int d0  = tid * 4;           // first of 4 columns owned by this thread

  // ---- pe[s, d0..d0+3] in registers, shared across the whole batch loop ----
  // angle = s * 10000^(-2*i/D) = s * exp2(-(2*i/D) * log2(10000))
  const float LOG2_10000 = 13.287712379549449f;
  v4f pe;
#pragma unroll
  for (int j = 0; j < 4; ++j) {
    const float fi  = (float)(d0 + j);
    const float ang = (float)s * exp2f(-(2.0f * fi / (float)PES_D) * LOG2_10000);
    pe[j] = ((d0 + j) & 1) ? cosf(ang) : sinf(ang);
  }

  const size_t row_off = (size_t)s * PES_D + (size_t)d0;     // elem offset in one batch
  const size_t bs4     = ((size_t)PES_S * PES_D) / 4;        // batch stride in v4f units

  const v4f* __restrict__ xin = (const v4f*)(x + row_off);
  v4f* __restrict__       o   = (v4f*)(out + row_off);

#if defined(HAVE_ASYNC_LDS)
  __shared__ v4f lbuf[2][THREADS];                           // 2 x 4 KiB double buffer

  // Prime the pipeline with batch 0.
  __builtin_amdgcn_global_load_async_to_lds_b128(
      (g4ptr)(xin), (l4ptr)(&lbuf[0][tid]), 0, 0);

  for (int b = 0; b < B; ++b) {
    if (b + 1 < B) {
      // Stage next batch row into the alternate buffer while we consume this one.
      __builtin_amdgcn_global_load_async_to_lds_b128(
          (g4ptr)(xin + (size_t)(b + 1) * bs4),
          (l4ptr)(&lbuf[(b + 1) & 1][tid]), 0, 0);
      wait_asynccnt_le<1>();   // async loads complete in order: oldest is done
    } else {
      wait_asynccnt_le<0>();
    }
    asm volatile("" ::: "memory");          // keep the ds read after the wait
    const v4f v = lbuf[b & 1][tid];         // ds_load_b128
    __builtin_nontemporal_store(v + pe, o + (size_t)b * bs4);
  }
#else
  // Fallback: plain streaming loads (still NT both ways).
  for (int b = 0; b < B; ++b) {
    const v4f v = __builtin_nontemporal_load(xin + (size_t)b * bs4);
    __builtin_nontemporal_store(v + pe, o + (size_t)b * bs4);
  }
#endif
}

extern "C" void kernel_launch(void* const* d_in, const int* in_sizes, int n_in,
                              void* d_out, int out_size, void* d_ws, size_t ws_size,
                              hipStream_t stream) {
  (void)n_in; (void)out_size; (void)d_ws; (void)ws_size;
  const float* x   = (const float*)d_in[0];
  float*       out = (float*)d_out;
  const int B = in_sizes[0] / (PES_S * PES_D);   // = 8 for the reference shapes
  dim3 grid(PES_S);
  dim3 block(THREADS);
  pe_add_kernel<<<grid, block, 0, stream>>>(x, out, B);
}